// attention_layer_51616916963412
// MI455X (gfx1250) — compile-verified
//
#include <hip/hip_runtime.h>
#include <hip/hip_bf16.h>

// GAT attention: B=32, N=1024, F=128.
//   s_src = feats @ a[:F], s_dst = feats @ a[F:]          (tiny GEMV -> WMMA f32)
//   e_ij  = leaky(s_src[i] + s_dst[j]); mask by adj; row softmax; zero non-edges
// Memory-bound: 128MB adj read + 128MB att write ~ 11us @ 23.3 TB/s.

#define B_    32
#define N_    1024
#define F_    128
#define ALPHA 0.2f
#define PADV  (-9e15f)

typedef __attribute__((ext_vector_type(2))) float v2f;
typedef __attribute__((ext_vector_type(8))) float v8f;

// ---------------------------------------------------------------------------
// Pass 1: per-row scores via V_WMMA_F32_16X16X4_F32.
// One wave per 16-row tile of feats (A-matrix), K-loop of 32 steps of K=4.
// A 16x4 f32 layout (ISA 7.12.2): lanes 0-15 -> M=lane, VGPR0=K0,VGPR1=K1;
//                                 lanes 16-31 -> M=lane-16, K2/K3.
// B 4x16 mirrored: lane = column N, VGPR0/1 = (K0,K1) for lanes 0-15,
//                  (K2,K3) for lanes 16-31. Column 0 = a1, column 1 = a2.
// D 16x16: VGPR r holds M=r (lanes 0-15) / M=8+r (lanes 16-31), N=lane&15.
//   => s_src lives in lanes {0,16}, s_dst in lanes {1,17}.
// B-fragment is built branchlessly: every lane loads from a clamped address
// (L0 hit) and lanes with column >= 2 select 0 via v_cndmask -> no EXEC
// save/restore inside the WMMA loop.
// ---------------------------------------------------------------------------
__global__ __launch_bounds__(256) void gat_scores_wmma(
    const float* __restrict__ feats, const float* __restrict__ avec,
    float* __restrict__ sSrc, float* __restrict__ sDst) {
  const int lane = threadIdx.x & 31;
  const int wave = threadIdx.x >> 5;
  const int tile = blockIdx.x * 8 + wave;          // 2048 tiles of 16 rows
  if (tile >= (B_ * N_) / 16) return;              // wave-uniform guard
  const int rowBase = tile * 16;
  const int n16 = lane & 15;                       // A: row M; B: column N
  const int hi  = lane >> 4;                       // K-pair select (0 -> K0K1, 1 -> K2K3)
  const bool live = (n16 < 2);                     // only B columns 0/1 nonzero

  const float* arow = feats + (size_t)(rowBase + n16) * F_ + 2 * hi;
  const float* brow = avec + (live ? n16 : 0) * F_ + 2 * hi;   // clamped, uniform-safe
  v8f c = {0.f, 0.f, 0.f, 0.f, 0.f, 0.f, 0.f, 0.f};

  #pragma unroll 4
  for (int k0 = 0; k0 < F_; k0 += 4) {
    v2f af = *(const v2f*)(arow + k0);             // 8B aligned (even float idx)
    v2f bl = *(const v2f*)(brow + k0);             // all lanes load (L0 hit)
    v2f bf;
    bf.x = live ? bl.x : 0.f;                      // v_cndmask, no divergence
    bf.y = live ? bl.y : 0.f;
    c = __builtin_amdgcn_wmma_f32_16x16x4_f32(
        /*neg_a=*/false, af, /*neg_b=*/false, bf,
        /*c_mod=*/(short)0, c, /*reuse_a=*/false, /*reuse_b=*/false);
  }

  if (live) {
    float* dst = (n16 == 0) ? sSrc : sDst;
    const int r0 = rowBase + hi * 8;
    #pragma unroll
    for (int r = 0; r < 8; ++r) dst[r0 + r] = c[r];
  }
}

// ---------------------------------------------------------------------------
// Pass 2: one wave32 per row i: 1024 logits = 8 x float4 per lane, kept in
// registers so adj is read exactly once. Wave-only max/sum reductions.
// ---------------------------------------------------------------------------
__device__ __forceinline__ float4 f4add(float s, float4 v) {
  float4 r; r.x = s + v.x; r.y = s + v.y; r.z = s + v.z; r.w = s + v.w; return r;
}

__global__ __launch_bounds__(256) void gat_softmax(
    const float* __restrict__ adj, const float* __restrict__ sSrc,
    const float* __restrict__ sDst, float* __restrict__ out) {
  const int lane = threadIdx.x & 31;
  const int wave = threadIdx.x >> 5;
  const int row  = blockIdx.x * 8 + wave;          // [0, B*N)
  if (row >= B_ * N_) return;                      // wave-uniform guard
  const int b = row >> 10;                         // N_ == 1024
  const size_t base = (size_t)row * N_;

  const float4* adjRow = (const float4*)(adj + base);
  const float4* sd     = (const float4*)(sDst + (size_t)b * N_);
  float4*       outRow = (float4*)(out + base);
  const float ssrc = sSrc[row];

  float4 em[8];
  float m = -3.4e38f;
  #pragma unroll
  for (int k = 0; k < 8; ++k) {
    const int j = lane + k * 32;                   // coalesced float4 index
    const float4 av = adjRow[j];
    const float4 sv = sd[j];
    float4 e = f4add(ssrc, sv);
    e.x = (e.x >= 0.f) ? e.x : ALPHA * e.x;
    e.y = (e.y >= 0.f) ? e.y : ALPHA * e.y;
    e.z = (e.z >= 0.f) ? e.z : ALPHA * e.z;
    e.w = (e.w >= 0.f) ? e.w : ALPHA * e.w;
    e.x = (av.x != 0.f) ? e.x : PADV;
    e.y = (av.y != 0.f) ? e.y : PADV;
    e.z = (av.z != 0.f) ? e.z : PADV;
    e.w = (av.w != 0.f) ? e.w : PADV;
    em[k] = e;
    m = fmaxf(m, fmaxf(fmaxf(e.x, e.y), fmaxf(e.z, e.w)));
  }
  #pragma unroll
  for (int o = 16; o > 0; o >>= 1) m = fmaxf(m, __shfl_xor(m, o, 32));

  float s = 0.f;
  #pragma unroll
  for (int k = 0; k < 8; ++k) {
    float4 e = em[k];
    e.x = __expf(e.x - m);                         // pads underflow to exact 0
    e.y = __expf(e.y - m);
    e.z = __expf(e.z - m);
    e.w = __expf(e.w - m);
    em[k] = e;
    s += (e.x + e.y) + (e.z + e.w);
  }
  #pragma unroll
  for (int o = 16; o > 0; o >>= 1) s += __shfl_xor(s, o, 32);
  const float inv = 1.0f / s;

  #pragma unroll
  for (int k = 0; k < 8; ++k) {
    float4 e = em[k];
    e.x *= inv; e.y *= inv; e.z *= inv; e.w *= inv;
    outRow[lane + k * 32] = e;
  }
}

// ---------------------------------------------------------------------------
extern "C" void kernel_launch(void* const* d_in, const int* in_sizes, int n_in,
                              void* d_out, int out_size, void* d_ws, size_t ws_size,
                              hipStream_t stream) {
  const float* adj   = (const float*)d_in[0];      // [B, N, N]
  const float* feats = (const float*)d_in[1];      // [B, N, F]
  const float* avec  = (const float*)d_in[2];      // [2F, 1] -> a1 | a2
  float* out  = (float*)d_out;                     // [B, N, N]
  float* sSrc = (float*)d_ws;                      // [B*N]
  float* sDst = sSrc + B_ * N_;                    // [B*N]

  // Pass 1: 2048 tiles, 8 waves/block -> 256 blocks.
  gat_scores_wmma<<<256, 256, 0, stream>>>(feats, avec, sSrc, sDst);
  // Pass 2: 32768 rows, 8 waves/block -> 4096 blocks.
  gat_softmax<<<4096, 256, 0, stream>>>(adj, sSrc, sDst, out);
}